// mLSTMLayer_39436389712571
// MI455X (gfx1250) — compile-verified
//
#include <hip/hip_runtime.h>
#include <hip/hip_bf16.h>
#include <math.h>

// Sizes (compile-time constants from the reference)
#define BATCH 64
#define FDIM  1024
#define HID   2048
#define NH    4
#define KS    4
#define BSZ   4
#define HD    512          // HID / NH
#define NB    512          // HID / BSZ
#define EPSV  1e-6f
#define KSCALE 0.04419417382415922f   // 1/sqrt(512)

typedef __attribute__((ext_vector_type(2))) float v2f;
typedef __attribute__((ext_vector_type(4))) float v4f;
typedef __attribute__((ext_vector_type(8))) float v8f;

// ---------------------------------------------------------------------------
// Generic fp32 GEMM:  C[M,N] = A[M,K] * B[K,N]   (row-major)
// Each wave computes a 16(M) x 64(N) strip using V_WMMA_F32_16X16X4_F32.
// A-fragment (16x4): lane 0-15 hold M=lane, VGPR0/1 = K {0,1} (lanes 0-15)
//                    or K {2,3} (lanes 16-31).
// B-fragment (4x16): VGPR0 = rows K0 (lanes 0-15) / K2 (lanes 16-31),
//                    VGPR1 = rows K1 / K3, column = lane&15.
// C/D (16x16): VGPR r, lane l -> row = r + 8*(l>=16), col = l&15.
// ---------------------------------------------------------------------------
__global__ void gemm_wmma(const float* __restrict__ A, const float* __restrict__ B,
                          float* __restrict__ Cout, int M, int N, int K)
{
    const int lane  = threadIdx.x & 31;
    const int wave  = (blockIdx.x * blockDim.x + threadIdx.x) >> 5;
    const int n64   = N >> 6;                 // 64-wide N tiles
    const int tm    = wave / n64;
    const int tn    = wave - tm * n64;
    if (tm * 16 >= M) return;

    const int l15   = lane & 15;
    const int khalf = lane >> 4;              // 0 or 1
    const int rowA  = tm * 16 + l15;

    v8f acc[4] = {};

    for (int kk = 0; kk < K; kk += 4) {
        const int ka = kk + khalf * 2;
        v2f a;
        a.x = A[(size_t)rowA * K + ka];
        a.y = A[(size_t)rowA * K + ka + 1];
#pragma unroll
        for (int t = 0; t < 4; ++t) {
            const int col = tn * 64 + t * 16 + l15;
            v2f bb;
            bb.x = B[(size_t)ka * N + col];
            bb.y = B[(size_t)(ka + 1) * N + col];
            acc[t] = __builtin_amdgcn_wmma_f32_16x16x4_f32(
                false, a, false, bb, (short)0, acc[t], false, false);
        }
    }

#pragma unroll
    for (int t = 0; t < 4; ++t) {
        const int col = tn * 64 + t * 16 + l15;
#pragma unroll
        for (int r = 0; r < 8; ++r) {
            const int row = tm * 16 + r + khalf * 8;
            Cout[(size_t)row * N + col] = acc[t][r];
        }
    }
}

// ---------------------------------------------------------------------------
// Per-batch kernel: conv-state shift + causal conv + silu, block-diagonal
// q/k/v projections (4x4 blocks), input/forget gate dots over qkv (6144),
// gate exponentials (m_new, i_p, f_p), n_new update and den scalar.
// ---------------------------------------------------------------------------
__global__ void gate_kernel(const float* __restrict__ up,
                            const float* __restrict__ conv_state,
                            const float* __restrict__ conv_w,
                            const float* __restrict__ conv_b,
                            const float* __restrict__ Wq,
                            const float* __restrict__ Wk,
                            const float* __restrict__ Wv,
                            const float* __restrict__ Wi,
                            const float* __restrict__ bi,
                            const float* __restrict__ Wf,
                            const float* __restrict__ bf,
                            const float* __restrict__ n_in,
                            const float* __restrict__ m_in,
                            float* __restrict__ csn_out,
                            float* __restrict__ nnew_out,
                            float* __restrict__ mnew_out,
                            float* __restrict__ cact_ws,
                            float* __restrict__ q_ws,
                            float* __restrict__ k_ws,
                            float* __restrict__ v_ws,
                            float* __restrict__ ip_ws,
                            float* __restrict__ fp_ws,
                            float* __restrict__ den_ws)
{
    const int b   = blockIdx.x;
    const int tid = threadIdx.x;

    __shared__ float s_x[HID];
    __shared__ float s_c[HID];
    __shared__ float s_q[HID];
    __shared__ float s_k[HID];
    __shared__ float s_v[HID];
    __shared__ float red_a[256];
    __shared__ float red_b[256];
    __shared__ float s_ip, s_fp, s_mn;

    // conv state shift + conv + silu
    for (int idx = tid; idx < HID; idx += 256) {
        float xv  = up[(size_t)b * (2 * HID) + idx];
        float cs1 = conv_state[((size_t)b * KS + 1) * HID + idx];
        float cs2 = conv_state[((size_t)b * KS + 2) * HID + idx];
        float cs3 = conv_state[((size_t)b * KS + 3) * HID + idx];
        csn_out[((size_t)b * KS + 0) * HID + idx] = cs1;
        csn_out[((size_t)b * KS + 1) * HID + idx] = cs2;
        csn_out[((size_t)b * KS + 2) * HID + idx] = cs3;
        csn_out[((size_t)b * KS + 3) * HID + idx] = xv;
        float cx = cs1 * conv_w[idx] + cs2 * conv_w[HID + idx] +
                   cs3 * conv_w[2 * HID + idx] + xv * conv_w[3 * HID + idx] +
                   conv_b[idx];
        float ca = cx / (1.0f + expf(-cx));       // silu
        s_x[idx] = xv;
        s_c[idx] = ca;
        cact_ws[(size_t)b * HID + idx] = ca;
    }
    __syncthreads();

    // block-diagonal projections (NB=512 blocks of 4x4)
    for (int idx = tid; idx < HID; idx += 256) {
        const int nb = idx >> 2;
        const int e  = idx & 3;
        float qv = 0.f, kv = 0.f, vv = 0.f;
#pragma unroll
        for (int d = 0; d < 4; ++d) {
            float cd = s_c[nb * 4 + d];
            float xd = s_x[nb * 4 + d];
            qv += cd * Wq[nb * 16 + d * 4 + e];
            kv += cd * Wk[nb * 16 + d * 4 + e];
            vv += xd * Wv[nb * 16 + d * 4 + e];
        }
        s_q[idx] = qv; s_k[idx] = kv; s_v[idx] = vv;
        q_ws[(size_t)b * HID + idx] = qv;
        k_ws[(size_t)b * HID + idx] = kv;
        v_ws[(size_t)b * HID + idx] = vv;
    }
    __syncthreads();

    // per-head gates + n update + den
    for (int h = 0; h < NH; ++h) {
        // i/f gate dot products over qkv (length 3*HID = 6144)
        float ai = 0.f, af = 0.f;
        for (int j = tid; j < 3 * HID; j += 256) {
            float qv = (j < HID) ? s_q[j] : (j < 2 * HID) ? s_k[j - HID] : s_v[j - 2 * HID];
            ai += qv * Wi[(size_t)j * NH + h];
            af += qv * Wf[(size_t)j * NH + h];
        }
        red_a[tid] = ai; red_b[tid] = af;
        __syncthreads();
        for (int s = 128; s > 0; s >>= 1) {
            if (tid < s) { red_a[tid] += red_a[tid + s]; red_b[tid] += red_b[tid + s]; }
            __syncthreads();
        }
        if (tid == 0) {
            float it = red_a[0] + bi[h];
            float ft = red_b[0] + bf[h];
            float lf = -log1pf(expf(-ft));              // -softplus(-f)
            float mo = m_in[b * NH + h];
            float mn = fmaxf(lf + mo, it);
            float ip = expf(it - mn);
            float fp = expf(lf + mo - mn);
            mnew_out[b * NH + h] = mn;
            ip_ws[b * NH + h] = ip;
            fp_ws[b * NH + h] = fp;
            s_ip = ip; s_fp = fp; s_mn = mn;
        }
        __syncthreads();
        const float ip = s_ip, fp = s_fp, mn = s_mn;

        // q.k_scaled and q.n dots + n_new write
        float aqk = 0.f, aqn = 0.f;
        for (int i = tid; i < HD; i += 256) {
            float qv = s_q[h * HD + i];
            float ks = s_k[h * HD + i] * KSCALE;
            float nv = n_in[(size_t)b * HID + h * HD + i];
            aqk += qv * ks;
            aqn += qv * nv;
            nnew_out[(size_t)b * HID + h * HD + i] = fp * nv + ip * ks;
        }
        red_a[tid] = aqk; red_b[tid] = aqn;
        __syncthreads();
        for (int s = 128; s > 0; s >>= 1) {
            if (tid < s) { red_a[tid] += red_a[tid + s]; red_b[tid] += red_b[tid + s]; }
            __syncthreads();
        }
        if (tid == 0) {
            float dd = fp * red_b[0] + ip * red_a[0];   // q^T n_new
            den_ws[b * NH + h] = fmaxf(fabsf(dd), expf(-mn));
        }
        __syncthreads();
    }
}

// ---------------------------------------------------------------------------
// Heavy kernel: C_new = f_p*C + i_p*(k_scaled v^T), fused with nom = q^T C_new.
// grid = (B*NH, 4 row-chunks), 128 threads; thread owns 4 columns (v4f);
// 512 MB stream touched exactly once -> non-temporal loads AND stores so the
// C stream does not sweep the 192 MB L2 (keeps GEMM weights resident).
// ---------------------------------------------------------------------------
__global__ void c_update_kernel(const float* __restrict__ C,
                                const float* __restrict__ q_ws,
                                const float* __restrict__ k_ws,
                                const float* __restrict__ v_ws,
                                const float* __restrict__ ip_ws,
                                const float* __restrict__ fp_ws,
                                float* __restrict__ Cnew,
                                float* __restrict__ nom_part)
{
    const int bh    = blockIdx.x;           // b*NH + h
    const int chunk = blockIdx.y;           // 0..3 (128 rows each)
    const int tid   = threadIdx.x;          // 0..127
    const int b     = bh >> 2;
    const int h     = bh & 3;

    __shared__ float s_coef[128];
    __shared__ float s_q[128];

    const float fp = fp_ws[bh];
    const float ip = ip_ws[bh];

    const float* qb = q_ws + (size_t)b * HID + h * HD;
    const float* kb = k_ws + (size_t)b * HID + h * HD;
    const float* vb = v_ws + (size_t)b * HID + h * HD;

    const int i0 = chunk * 128;
    s_q[tid]    = qb[i0 + tid];
    s_coef[tid] = ip * kb[i0 + tid] * KSCALE;
    __syncthreads();

    const int j0 = tid * 4;
    v4f vj = *(const v4f*)(vb + j0);
    v4f nom = {0.f, 0.f, 0.f, 0.f};

    const size_t base = (size_t)bh * HD * HD;
#pragma unroll 8
    for (int i = 0; i < 128; ++i) {
        const size_t off = base + (size_t)(i0 + i) * HD + j0;
        // prefetch a few rows ahead (global_prefetch_b8)
        if (i + 8 < 128)
            __builtin_prefetch((const void*)(C + off + 8 * HD), 0, 0);
        v4f c = __builtin_nontemporal_load((const v4f*)(C + off));
        const float co = s_coef[i];
        v4f cn = fp * c + co * vj;
        __builtin_nontemporal_store(cn, (v4f*)(Cnew + off));
        const float qi = s_q[i];
        nom += qi * cn;
    }
    *(v4f*)(nom_part + (size_t)chunk * (BATCH * NH * HD) + (size_t)bh * HD + j0) = nom;
}

// ---------------------------------------------------------------------------
// h_tilde -> layernorm (mean/var over HD) -> + skip*conv_act -> * silu(z)
// ---------------------------------------------------------------------------
__global__ void norm_kernel(const float* __restrict__ nom_part,
                            const float* __restrict__ den_ws,
                            const float* __restrict__ cact_ws,
                            const float* __restrict__ up,
                            const float* __restrict__ norm_scale,
                            const float* __restrict__ skip,
                            float* __restrict__ hfull)
{
    const int bh  = blockIdx.x;
    const int tid = threadIdx.x;
    const int b   = bh >> 2;
    const int h   = bh & 3;

    __shared__ float s_h[HD];
    __shared__ float red_a[256];
    __shared__ float red_b[256];
    __shared__ float s_mu, s_rs;

    const float den = den_ws[bh] + EPSV;
    float lsum = 0.f, lsq = 0.f;
    for (int j = tid; j < HD; j += 256) {
        const size_t o = (size_t)bh * HD + j;
        float nv = nom_part[o] + nom_part[(size_t)(BATCH * NH * HD) + o] +
                   nom_part[(size_t)2 * (BATCH * NH * HD) + o] +
                   nom_part[(size_t)3 * (BATCH * NH * HD) + o];
        float ht = nv / den;
        s_h[j] = ht;
        lsum += ht;
        lsq  += ht * ht;
    }
    red_a[tid] = lsum; red_b[tid] = lsq;
    __syncthreads();
    for (int s = 128; s > 0; s >>= 1) {
        if (tid < s) { red_a[tid] += red_a[tid + s]; red_b[tid] += red_b[tid + s]; }
        __syncthreads();
    }
    if (tid == 0) {
        float mu  = red_a[0] * (1.0f / HD);
        float var = red_b[0] * (1.0f / HD) - mu * mu;
        s_mu = mu;
        s_rs = rsqrtf(var + EPSV);
    }
    __syncthreads();
    const float mu = s_mu, rs = s_rs;

    for (int j = tid; j < HD; j += 256) {
        float hn = (s_h[j] - mu) * rs * norm_scale[h * HD + j];
        const size_t idx = (size_t)b * HID + h * HD + j;
        float hv = hn + skip[h * HD + j] * cact_ws[idx];
        float z  = up[(size_t)b * (2 * HID) + HID + h * HD + j];
        hv *= z / (1.0f + expf(-z));              // * silu(z)
        hfull[idx] = hv;
    }
}

// ---------------------------------------------------------------------------
extern "C" void kernel_launch(void* const* d_in, const int* in_sizes, int n_in,
                              void* d_out, int out_size, void* d_ws, size_t ws_size,
                              hipStream_t stream)
{
    const float* inputs     = (const float*)d_in[0];
    const float* C          = (const float*)d_in[1];
    const float* n_in_p     = (const float*)d_in[2];
    const float* m_in_p     = (const float*)d_in[3];
    const float* conv_state = (const float*)d_in[4];
    const float* W_up       = (const float*)d_in[5];
    const float* conv_w     = (const float*)d_in[6];
    const float* conv_b     = (const float*)d_in[7];
    const float* Wq         = (const float*)d_in[8];
    const float* Wk         = (const float*)d_in[9];
    const float* Wv         = (const float*)d_in[10];
    const float* Wi         = (const float*)d_in[11];
    const float* bi         = (const float*)d_in[12];
    const float* Wf         = (const float*)d_in[13];
    const float* bf         = (const float*)d_in[14];
    const float* norm_scale = (const float*)d_in[15];
    const float* skip       = (const float*)d_in[16];
    const float* W_down     = (const float*)d_in[17];

    float* out  = (float*)d_out;
    float* y    = out;                                       // 64*1024
    float* Cnew = out + (size_t)BATCH * FDIM;                // 64*4*512*512
    float* nnew = Cnew + (size_t)BATCH * NH * HD * HD;       // 64*4*512
    float* mnew = nnew + (size_t)BATCH * HID;                // 64*4
    float* csn  = mnew + (size_t)BATCH * NH;                 // 64*4*2048

    float* ws      = (float*)d_ws;
    float* up_ws   = ws;                       // 64*4096   = 262144
    float* cact_ws = up_ws + 262144;           // 64*2048   = 131072
    float* q_ws    = cact_ws + 131072;         // 131072
    float* k_ws    = q_ws + 131072;            // 131072
    float* v_ws    = k_ws + 131072;            // 131072
    float* ip_ws   = v_ws + 131072;            // 256
    float* fp_ws   = ip_ws + 256;              // 256
    float* den_ws  = fp_ws + 256;              // 256
    float* nom_ws  = den_ws + 256;             // 4*64*4*512 = 524288
    float* hfull   = nom_ws + 524288;          // 131072

    // 1) up = inputs @ W_up   (64x1024 @ 1024x4096): 256 waves -> 32 blocks
    gemm_wmma<<<32, 256, 0, stream>>>(inputs, W_up, up_ws, BATCH, 2 * HID, FDIM);

    // 2) conv + blockdiag qkv + gates + n_new + den
    gate_kernel<<<BATCH, 256, 0, stream>>>(up_ws, conv_state, conv_w, conv_b,
                                           Wq, Wk, Wv, Wi, bi, Wf, bf,
                                           n_in_p, m_in_p,
                                           csn, nnew, mnew,
                                           cact_ws, q_ws, k_ws, v_ws,
                                           ip_ws, fp_ws, den_ws);

    // 3) C_new + fused nom (the 512 MB streaming kernel, NT hints)
    c_update_kernel<<<dim3(BATCH * NH, 4), 128, 0, stream>>>(C, q_ws, k_ws, v_ws,
                                                             ip_ws, fp_ws,
                                                             Cnew, nom_ws);

    // 4) normalize + skip + silu(z) gate
    norm_kernel<<<BATCH * NH, 256, 0, stream>>>(nom_ws, den_ws, cact_ws, up_ws,
                                                norm_scale, skip, hfull);

    // 5) y = h @ W_down   (64x2048 @ 2048x1024): 64 waves -> 8 blocks
    gemm_wmma<<<8, 256, 0, stream>>>(hfull, W_down, y, BATCH, FDIM, HID);
}